// MoE_81432579932270
// MI455X (gfx1250) — compile-verified
//
#include <hip/hip_runtime.h>

// ---------------------------------------------------------------------------
// MoE (sigmoid router, top-4 of 32 experts, per-expert ReLU MLP) for gfx1250.
// fp32 end-to-end using V_WMMA_F32_16X16X4_F32. Problem is small (4.3 GFLOP
// sparse, 32MB resident -> fits the 192MB L2); the limiter is per-tile weight
// re-reads from L2, so we use 64-token M-tiles (4x B-fragment reuse) with a
// 163KB dynamic-LDS staging buffer (WGP has 320KB). The X-tile gather uses
// GLOBAL_LOAD_ASYNC_TO_LDS_B128 (ASYNCcnt path, no VGPR round-trip).
// ---------------------------------------------------------------------------

#define NTOK 4096
#define DMODEL 512
#define NEXP 32
#define HID 128
#define TOPK 4

#define MTILE 64            // tokens per expert block (4 x 16-row WMMA tiles)
#define XSTR 516            // X-tile LDS row stride (516 % 64 == 4 -> no bank conflicts)
#define HSTR 132            // h-tile LDS row stride (132 % 64 == 4)
// dynamic LDS layout (floats): x[MTILE*XSTR] | h[MTILE*HSTR] | g[MTILE] | t[MTILE]
#define SMEM_FLOATS (MTILE * XSTR + MTILE * HSTR + MTILE + MTILE)
#define SMEM_BYTES (SMEM_FLOATS * 4)

typedef __attribute__((ext_vector_type(2))) float v2f;
typedef __attribute__((ext_vector_type(8))) float v8f;

// ---------------------------------------------------------------------------
// 0) zero output + expert counts (harness poisons buffers with 0xAA)
// ---------------------------------------------------------------------------
__global__ void moe_zero_kernel(float4* __restrict__ y4, int* __restrict__ counts) {
    int i = blockIdx.x * 256 + threadIdx.x;
    if (i < NTOK * DMODEL / 4) y4[i] = make_float4(0.f, 0.f, 0.f, 0.f);
    if (i < NEXP) counts[i] = 0;
}

// ---------------------------------------------------------------------------
// 1) router: one wave32 per token. lane e owns expert e's dot product.
//    serial argmax x4 reproduces jax.lax.top_k tie-breaking (first index).
// ---------------------------------------------------------------------------
__global__ void __launch_bounds__(128)
moe_router_kernel(const float* __restrict__ x, const float* __restrict__ esel,
                  int* __restrict__ gidx, float* __restrict__ gval,
                  int* __restrict__ counts) {
    const int wave = threadIdx.x >> 5;
    const int lane = threadIdx.x & 31;
    const int token = blockIdx.x * 4 + wave;

    __shared__ float lds_x[4][DMODEL];
    __shared__ float lds_s[4][NEXP];

    for (int d = lane; d < DMODEL; d += 32)
        lds_x[wave][d] = x[token * DMODEL + d];
    __syncthreads();

    const float* w = esel + lane * DMODEL;   // lane = expert id
    float s = 0.0f;
    #pragma unroll 8
    for (int d = 0; d < DMODEL; ++d)
        s += lds_x[wave][d] * w[d];
    s = 1.0f / (1.0f + __expf(-s));          // sigmoid gate
    lds_s[wave][lane] = s;
    __syncthreads();

    if (lane == 0) {
        #pragma unroll
        for (int k = 0; k < TOPK; ++k) {
            int best = 0;
            float bv = -1e30f;
            for (int e = 0; e < NEXP; ++e) {
                float v = lds_s[wave][e];
                if (v > bv) { bv = v; best = e; }
            }
            gidx[token * TOPK + k] = best;
            gval[token * TOPK + k] = bv;
            lds_s[wave][best] = -1e30f;
            atomicAdd(&counts[best], 1);
        }
    }
}

// ---------------------------------------------------------------------------
// 2) exclusive scan over 32 expert counts
// ---------------------------------------------------------------------------
__global__ void moe_scan_kernel(const int* __restrict__ counts,
                                int* __restrict__ offsets,
                                int* __restrict__ cursor) {
    if (threadIdx.x == 0 && blockIdx.x == 0) {
        int acc = 0;
        for (int e = 0; e < NEXP; ++e) {
            offsets[e] = acc;
            cursor[e]  = acc;
            acc += counts[e];
        }
        offsets[NEXP] = acc;
    }
}

// ---------------------------------------------------------------------------
// 3) scatter (token, k) -> permuted per-expert slot
// ---------------------------------------------------------------------------
__global__ void moe_scatter_kernel(const int* __restrict__ gidx,
                                   const float* __restrict__ gval,
                                   int* __restrict__ cursor,
                                   int* __restrict__ perm,
                                   float* __restrict__ pgate) {
    int i = blockIdx.x * 256 + threadIdx.x;   // i in [0, N*K)
    if (i >= NTOK * TOPK) return;
    int e = gidx[i];
    int pos = atomicAdd(&cursor[e], 1);
    perm[pos]  = i >> 2;                      // token id
    pgate[pos] = gval[i];
}

// ---------------------------------------------------------------------------
// 4) per-expert MLP over 64-token tiles. block = 256 threads = 8 wave32s.
//    Phase A: h[64,128] = relu(Xg @ W1[e]) * gate
//      wave w -> h column tile n0=16w; B fragment reused across 4 M-tiles.
//    Phase B: y[tok,:] += h @ W2[e]
//      wave w -> 4 column tiles n0=64w+16t; each B fragment reused 4x.
//    A/B fragments per ISA 32-bit 16x4 layout: lane = {M|N} + 16*(K>=2),
//    VGPR j holds K = j + 2*(lane>=16). C/D: VGPR r, lanes>=16 -> M=r+8.
// ---------------------------------------------------------------------------
__global__ void __launch_bounds__(256)
moe_expert_kernel(const float* __restrict__ x,
                  const float* __restrict__ w1,
                  const float* __restrict__ w2,
                  const int* __restrict__ offsets,
                  const int* __restrict__ perm,
                  const float* __restrict__ pgate,
                  float* __restrict__ y) {
    const int e   = blockIdx.x;
    const int beg = offsets[e];
    const int end = offsets[e + 1];
    const int tileStart = beg + blockIdx.y * MTILE;
    if (tileStart >= end) return;
    const int nTok = min(MTILE, end - tileStart);

    extern __shared__ float smem[];
    float* lds_x = smem;                                  // [MTILE * XSTR]
    float* lds_h = smem + MTILE * XSTR;                   // [MTILE * HSTR]
    float* lds_g = smem + MTILE * XSTR + MTILE * HSTR;    // [MTILE]
    int*   lds_t = (int*)(lds_g + MTILE);                 // [MTILE]

    const int tid = threadIdx.x;
    if (tid < MTILE) {
        const bool valid = tid < nTok;
        lds_t[tid] = valid ? perm[tileStart + tid] : 0;   // pad rows -> token 0
        lds_g[tid] = valid ? pgate[tileStart + tid] : 0.0f; // pad gate 0 -> h row 0
    }
    __syncthreads();

    // gather 64 token rows straight into LDS via the async DMA path:
    // GLOBAL_LOAD_ASYNC_TO_LDS_B128, 16B/lane, ASYNCcnt-tracked, no VGPR
    // round-trip. LDS row stride 2064B keeps every store 16B aligned.
    for (int idx = tid; idx < MTILE * (DMODEL / 4); idx += 256) {
        const int r  = idx >> 7;              // DMODEL/4 == 128
        const int c4 = idx & 127;
        // low 32 bits of a generic shared pointer == LDS byte offset
        const unsigned int ldsoff =
            (unsigned int)(size_t)(lds_x + r * XSTR + c4 * 4);
        const float* gsrc = x + (size_t)lds_t[r] * DMODEL + c4 * 4;
        asm volatile("global_load_async_to_lds_b128 %0, %1, off"
                     :: "v"(ldsoff), "v"(gsrc)
                     : "memory");
    }
    asm volatile("s_wait_asynccnt 0x0" ::: "memory");
    __syncthreads();

    const int wave = tid >> 5;
    const int lane = tid & 31;
    const int hi   = lane >> 4;               // K half-select
    const int lo   = lane & 15;               // M (A) or N (B/C/D)

    const float* W2 = w2 + (size_t)e * HID * DMODEL;
    // warm the W2 panel for Phase B while Phase A runs
    __builtin_prefetch(W2 + wave * 64 + lane * 8, 0, 0);

    // ---- Phase A: h = relu(Xg @ W1[e]) * gate  (64x128, wave -> 16 cols) --
    {
        const int n0 = wave * 16;
        const float* W1 = w1 + (size_t)e * DMODEL * HID + n0;
        v8f acc0 = {}, acc1 = {}, acc2 = {}, acc3 = {};
        for (int kk = 0; kk < DMODEL; kk += 4) {
            const int k0 = kk + 2 * hi;       // even -> 8B-aligned LDS loads
            v2f b;
            b.x = W1[(k0    ) * HID + lo];
            b.y = W1[(k0 + 1) * HID + lo];
            v2f a0 = *(const v2f*)(lds_x + (lo     ) * XSTR + k0);
            v2f a1 = *(const v2f*)(lds_x + (lo + 16) * XSTR + k0);
            v2f a2 = *(const v2f*)(lds_x + (lo + 32) * XSTR + k0);
            v2f a3 = *(const v2f*)(lds_x + (lo + 48) * XSTR + k0);
            acc0 = __builtin_amdgcn_wmma_f32_16x16x4_f32(false, a0, false, b, (short)0, acc0, false, false);
            acc1 = __builtin_amdgcn_wmma_f32_16x16x4_f32(false, a1, false, b, (short)0, acc1, false, false);
            acc2 = __builtin_amdgcn_wmma_f32_16x16x4_f32(false, a2, false, b, (short)0, acc2, false, false);
            acc3 = __builtin_amdgcn_wmma_f32_16x16x4_f32(false, a3, false, b, (short)0, acc3, false, false);
        }
        #pragma unroll
        for (int mt = 0; mt < 4; ++mt) {
            const v8f acc = (mt == 0) ? acc0 : (mt == 1) ? acc1 : (mt == 2) ? acc2 : acc3;
            #pragma unroll
            for (int r = 0; r < 8; ++r) {
                const int m = mt * 16 + r + 8 * hi;
                const float g = lds_g[m];
                lds_h[m * HSTR + n0 + lo] = fmaxf(acc[r], 0.0f) * g;
            }
        }
    }
    __syncthreads();

    // ---- Phase B: y[tok,:] += h @ W2[e]  (64x512, wave -> 4 col tiles) ----
    #pragma unroll
    for (int t = 0; t < 4; ++t) {
        const int n0 = wave * 64 + t * 16;
        v8f acc0 = {}, acc1 = {}, acc2 = {}, acc3 = {};
        for (int kk = 0; kk < HID; kk += 4) {
            const int k0 = kk + 2 * hi;
            v2f b;
            b.x = W2[(k0    ) * DMODEL + n0 + lo];
            b.y = W2[(k0 + 1) * DMODEL + n0 + lo];
            v2f a0 = *(const v2f*)(lds_h + (lo     ) * HSTR + k0);
            v2f a1 = *(const v2f*)(lds_h + (lo + 16) * HSTR + k0);
            v2f a2 = *(const v2f*)(lds_h + (lo + 32) * HSTR + k0);
            v2f a3 = *(const v2f*)(lds_h + (lo + 48) * HSTR + k0);
            acc0 = __builtin_amdgcn_wmma_f32_16x16x4_f32(false, a0, false, b, (short)0, acc0, false, false);
            acc1 = __builtin_amdgcn_wmma_f32_16x16x4_f32(false, a1, false, b, (short)0, acc1, false, false);
            acc2 = __builtin_amdgcn_wmma_f32_16x16x4_f32(false, a2, false, b, (short)0, acc2, false, false);
            acc3 = __builtin_amdgcn_wmma_f32_16x16x4_f32(false, a3, false, b, (short)0, acc3, false, false);
        }
        #pragma unroll
        for (int mt = 0; mt < 4; ++mt) {
            const v8f acc = (mt == 0) ? acc0 : (mt == 1) ? acc1 : (mt == 2) ? acc2 : acc3;
            #pragma unroll
            for (int r = 0; r < 8; ++r) {
                const int m = mt * 16 + r + 8 * hi;
                if (m < nTok)
                    atomicAdd(&y[(size_t)lds_t[m] * DMODEL + n0 + lo], acc[r]);
            }
        }
    }
}

// ---------------------------------------------------------------------------
// launch
// ---------------------------------------------------------------------------
extern "C" void kernel_launch(void* const* d_in, const int* in_sizes, int n_in,
                              void* d_out, int out_size, void* d_ws, size_t ws_size,
                              hipStream_t stream) {
    const float* x    = (const float*)d_in[0];   // [N, D]
    const float* esel = (const float*)d_in[1];   // [E, D]
    const float* w1   = (const float*)d_in[2];   // [E, D, H]
    const float* w2   = (const float*)d_in[3];   // [E, H, D]
    float* y = (float*)d_out;                    // [N, D]

    // workspace layout (~260 KB)
    char* ws = (char*)d_ws;
    int*   gidx    = (int*)  (ws + 0);           // N*K ints   (64 KB)
    float* gval    = (float*)(ws + (64 << 10));  // N*K floats (64 KB)
    int*   perm    = (int*)  (ws + (128 << 10)); // N*K ints   (64 KB)
    float* pgate   = (float*)(ws + (192 << 10)); // N*K floats (64 KB)
    int*   counts  = (int*)  (ws + (256 << 10));          // E ints
    int*   offsets = (int*)  (ws + (256 << 10) + 256);    // E+1 ints
    int*   cursor  = (int*)  (ws + (256 << 10) + 512);    // E ints

    // allow >64KB dynamic LDS for the expert kernel (WGP has 320KB)
    (void)hipFuncSetAttribute((const void*)moe_expert_kernel,
                              hipFuncAttributeMaxDynamicSharedMemorySize,
                              SMEM_BYTES);

    // 0) zero y + counts
    moe_zero_kernel<<<(NTOK * DMODEL / 4 + 255) / 256, 256, 0, stream>>>(
        (float4*)y, counts);
    // 1) route
    moe_router_kernel<<<NTOK / 4, 128, 0, stream>>>(x, esel, gidx, gval, counts);
    // 2) scan
    moe_scan_kernel<<<1, 32, 0, stream>>>(counts, offsets, cursor);
    // 3) scatter / permute
    moe_scatter_kernel<<<(NTOK * TOPK + 255) / 256, 256, 0, stream>>>(
        gidx, gval, cursor, perm, pgate);
    // 4) per-expert MLP (64-token tiles; idle tiles exit immediately)
    dim3 grid(NEXP, NTOK / MTILE);
    moe_expert_kernel<<<grid, 256, SMEM_BYTES, stream>>>(
        x, w1, w2, offsets, perm, pgate, y);
}